// MicroscopyRadon_35751307771955
// MI455X (gfx1250) — compile-verified
//
#include <hip/hip_runtime.h>

typedef __attribute__((ext_vector_type(2))) float v2f;
typedef __attribute__((ext_vector_type(8))) float v8f;

#define IMG     512
#define NANG    180
#define NBATCH  2
#define WCHUNK  64            // w columns per block
#define HGROUPS 4             // h strips per block
#define HPER    (IMG / HGROUPS) // 128 h samples per thread

// Radon forward projection, grid-driven bilinear gather (matches jax reference).
// Block: 256 threads = 64 w-lanes x 4 h-groups. Each thread sums 128 bilinear
// samples; partials reduced across h-groups with V_WMMA_F32_16X16X4_F32.
__global__ __launch_bounds__(256)
void MicroscopyRadon_kernel(const float* __restrict__ img,
                            const float* __restrict__ angles,
                            float* __restrict__ out) {
    __shared__ float P[HGROUPS][WCHUNK];

    const int tid = threadIdx.x;
    const int wl  = tid & (WCHUNK - 1);   // w lane within block
    const int hg  = tid >> 6;             // h group 0..3

    int bb = blockIdx.x;
    const int wc = bb & 7;  bb >>= 3;     // w chunk 0..7
    const int n  = bb % NANG;             // angle
    const int b  = bb / NANG;             // batch

    const int w = wc * WCHUNK + wl;

    // angle in degrees -> radians; cos/sin once per thread
    const float th = angles[n] * 0.017453292519943295f;
    float s, c;
    sincosf(th, &s, &c);

    // ix = (c*X - s*Y + 1) * 255.5 ; iy = (s*X + c*Y + 1) * 255.5
    const float cs = c * 255.5f;
    const float ss = s * 255.5f;
    const float Xw = fmaf((float)w, 2.0f / 511.0f, -1.0f);
    const float ax = fmaf(cs, Xw, 255.5f);   // x-coord base for this w
    const float ay = fmaf(ss, Xw, 255.5f);   // y-coord base for this w

    const float* __restrict__ im = img + b * (IMG * IMG);

    float acc = 0.0f;
    const int h0 = hg * HPER;
    #pragma unroll 4
    for (int i = 0; i < HPER; ++i) {
        const float Yh = fmaf((float)(h0 + i), 2.0f / 511.0f, -1.0f);
        const float ix = fmaf(-ss, Yh, ax);
        const float iy = fmaf(cs, Yh, ay);

        const float x0f = floorf(ix);
        const float y0f = floorf(iy);
        const float fx = ix - x0f;
        const float fy = iy - y0f;
        const int x0 = (int)x0f;
        const int y0 = (int)y0f;
        const int x1 = x0 + 1;
        const int y1 = y0 + 1;

        // zero-padding: separable validity folded into the weights
        const float wx0 = ((unsigned)x0 < 512u) ? (1.0f - fx) : 0.0f;
        const float wx1 = ((unsigned)x1 < 512u) ? fx          : 0.0f;
        const float wy0 = ((unsigned)y0 < 512u) ? (1.0f - fy) : 0.0f;
        const float wy1 = ((unsigned)y1 < 512u) ? fy          : 0.0f;

        const int x0c = min(max(x0, 0), 511);
        const int x1c = min(max(x1, 0), 511);
        const int r0  = min(max(y0, 0), 511) << 9;
        const int r1  = min(max(y1, 0), 511) << 9;

        const float v00 = im[r0 + x0c];
        const float v01 = im[r0 + x1c];
        const float v10 = im[r1 + x0c];
        const float v11 = im[r1 + x1c];

        const float top = fmaf(wx1, v01, wx0 * v00);
        const float bot = fmaf(wx1, v11, wx0 * v10);
        acc = fmaf(wy0, top, acc);
        acc = fmaf(wy1, bot, acc);
    }

    P[hg][wl] = acc;
    __syncthreads();

    // Cross-h-group reduction via f32 WMMA: D = ones(16x4) * P(4x16) + 0.
    // Every row of D equals the column sums; row M=0 lives in C/D VGPR0,
    // lanes 0..15. Waves 0..3 each handle one 16-wide w chunk (EXEC all-1s
    // under this wave-uniform branch, as WMMA requires).
    if (tid < 128) {
        const int wave = tid >> 5;
        const int lane = tid & 31;
        const int n16  = lane & 15;
        const int kb   = (lane >> 4) * 2;     // B layout: VGPR0=K{0|2}, VGPR1=K{1|3}
        const int wcol = wave * 16 + n16;

        v2f Bm;
        Bm.x = P[kb][wcol];
        Bm.y = P[kb + 1][wcol];
        v2f Am;
        Am.x = 1.0f;
        Am.y = 1.0f;
        v8f Cm = {};
        Cm = __builtin_amdgcn_wmma_f32_16x16x4_f32(
                 /*neg_a=*/false, Am, /*neg_b=*/false, Bm,
                 /*c_mod=*/(short)0, Cm, /*reuse_a=*/false, /*reuse_b=*/false);

        if (lane < 16) {
            out[(b * NANG + n) * IMG + wc * WCHUNK + wcol] = Cm[0];
        }
    }
}

extern "C" void kernel_launch(void* const* d_in, const int* in_sizes, int n_in,
                              void* d_out, int out_size, void* d_ws, size_t ws_size,
                              hipStream_t stream) {
    const float* img    = (const float*)d_in[0];   // [2,1,512,512] f32
    const float* angles = (const float*)d_in[1];   // [180] f32 degrees
    float* out          = (float*)d_out;           // [2,180,512] f32

    dim3 grid(NBATCH * NANG * (IMG / WCHUNK));     // 2*180*8 = 2880 blocks
    dim3 block(256);
    hipLaunchKernelGGL(MicroscopyRadon_kernel, grid, block, 0, stream,
                       img, angles, out);
}